// Model_40157944218444
// MI455X (gfx1250) — compile-verified
//
#include <hip/hip_runtime.h>

// Problem constants from the reference.
#define F_NUM   2048      // FEATURE_NUM
#define OUT_NUM 256       // OUTPUT_NUM

typedef __attribute__((ext_vector_type(2))) float v2f;
typedef __attribute__((ext_vector_type(4))) float v4f;
typedef __attribute__((ext_vector_type(8))) float v8f;

// ---------------------------------------------------------------------------
// Kernel 0: zero the F_NUM-float partial-sum accumulator in workspace.
// ---------------------------------------------------------------------------
__global__ void zero_ws_kernel(float* __restrict__ sacc) {
    int i = blockIdx.x * blockDim.x + threadIdx.x;
    if (i < F_NUM) sacc[i] = 0.0f;
}

// ---------------------------------------------------------------------------
// Kernel 1 (bandwidth-dominant): sacc[i] = sum_n adj[n] * fm[n, i]
// feature_map (409.6 MB) is streamed exactly once and is larger than the
// 192 MB L2, so loads are non-temporal (TH=NT b128) to avoid cache thrash.
// Grid: (2 column slabs, 256 row chunks) = 512 blocks x 256 threads
// (4096 waves in flight). Each thread owns a float4 column strip,
// accumulates in registers over its row chunk, then does 4 atomicAdds.
// ---------------------------------------------------------------------------
#define THREADS1 256

__global__ __launch_bounds__(THREADS1)
void colsum_kernel(const float* __restrict__ adj,
                   const float* __restrict__ fm,
                   float* __restrict__ sacc,
                   int n_nodes) {
    const int col = (blockIdx.x * THREADS1 + threadIdx.x) * 4;   // 0..F_NUM-4
    const int rows_per = (n_nodes + gridDim.y - 1) / gridDim.y;
    const int n0 = blockIdx.y * rows_per;
    int n1 = n0 + rows_per;
    if (n1 > n_nodes) n1 = n_nodes;

    v4f acc = {0.f, 0.f, 0.f, 0.f};
    #pragma unroll 4
    for (int n = n0; n < n1; ++n) {
        const float w = adj[n];
        const v4f f = __builtin_nontemporal_load(
            (const v4f*)(fm + (size_t)n * F_NUM + col));
        acc += w * f;
    }
    atomicAdd(&sacc[col + 0], acc.x);
    atomicAdd(&sacc[col + 1], acc.y);
    atomicAdd(&sacc[col + 2], acc.z);
    atomicAdd(&sacc[col + 3], acc.w);
}

// ---------------------------------------------------------------------------
// Kernel 2: agg[i] = A[i] * sacc[i]   (staged in LDS), then
//           out[1,256] = agg[1,2048] @ B[2048,256]
// via V_WMMA_F32_16X16X4_F32 with M padded 1 -> 16 (rows 1..15 zero).
//
// f32 16x4 A layout (ISA 7.12.2): lanes 0-15 hold M=0..15; VGPR0 = {K=0 | K=2
// in lane halves}, VGPR1 = {K=1 | K=3}.  Only M=0 is live -> lane 0 carries
// {agg[k],agg[k+1]}, lane 16 carries {agg[k+2],agg[k+3]}, all else 0.
// The agg pair is loaded UNconditionally (ds_load_b64, uniform address per
// half-wave) and masked with v_cndmask — no exec-mask round trips in the loop.
// f32 4x16 B layout (row striped across lanes, mirrored K halves):
// VGPR0 lane L<16 = B[k+0, n], lane L>=16 = B[k+2, n]; VGPR1 = K+1 / K+3.
// D row M=0 = C-VGPR0 lanes 0-15 (N = lane).
// ---------------------------------------------------------------------------
__global__ __launch_bounds__(512)
void wmma_epilogue_kernel(const float* __restrict__ sacc,
                          const float* __restrict__ A,
                          const float* __restrict__ B,
                          float* __restrict__ out) {
    __shared__ float agg[F_NUM];
    for (int i = threadIdx.x; i < F_NUM; i += 512)
        agg[i] = A[i] * sacc[i];
    __syncthreads();

    const int lane = threadIdx.x & 31;
    const int wave = threadIdx.x >> 5;          // 16 waves -> 16 N-tiles
    const int n    = wave * 16 + (lane & 15);   // output column
    const int kk   = (lane >> 4) * 2;           // 0 for lanes 0-15, 2 for 16-31
    const bool head = (lane & 15) == 0;         // lanes holding M=0 of A

    v8f c = {};
    #pragma unroll 8
    for (int k = 0; k < F_NUM; k += 4) {
        // Unconditional LDS load (all lanes), then mask: 1 ds_load_b64 + 2 cndmask.
        const v2f av = *(const v2f*)(agg + k + kk);   // 8B aligned (k+kk even)
        v2f a;
        a.x = head ? av.x : 0.0f;
        a.y = head ? av.y : 0.0f;
        v2f b;
        b.x = __builtin_nontemporal_load(B + (size_t)(k + kk)     * OUT_NUM + n);
        b.y = __builtin_nontemporal_load(B + (size_t)(k + kk + 1) * OUT_NUM + n);
        // 8 args: (neg_a, A, neg_b, B, c_mod, C, reuse_a, reuse_b)
        c = __builtin_amdgcn_wmma_f32_16x16x4_f32(
                false, a, false, b, (short)0, c, false, false);
    }

    // Row M=0 of D: C-VGPR0 across lanes 0..15 (N = lane).
    if (lane < 16) out[wave * 16 + lane] = c[0];
}

// ---------------------------------------------------------------------------
// Launch. Inputs (setup_inputs order):
//   d_in[0] node_adjacency [1,N] f32     d_in[1] feature_map [N,F] f32
//   d_in[2] A [F,1] f32                  d_in[3] B [F,256] f32
//   d_in[4] fe_mask_rate (==0, identity dropout -> ignored)
// Output: d_out = [1,256] f32.  Workspace: F_NUM floats (8 KB) of partials.
// ---------------------------------------------------------------------------
extern "C" void kernel_launch(void* const* d_in, const int* in_sizes, int n_in,
                              void* d_out, int out_size, void* d_ws, size_t ws_size,
                              hipStream_t stream) {
    const float* adj = (const float*)d_in[0];
    const float* fm  = (const float*)d_in[1];
    const float* A   = (const float*)d_in[2];
    const float* B   = (const float*)d_in[3];
    float* out  = (float*)d_out;
    float* sacc = (float*)d_ws;          // F_NUM floats
    const int n_nodes = in_sizes[0];     // 50000

    zero_ws_kernel<<<(F_NUM + 255) / 256, 256, 0, stream>>>(sacc);

    const int col_slabs = F_NUM / (THREADS1 * 4);  // 2
    const int nsplit    = 256;                     // 512 blocks, 4096 waves
    colsum_kernel<<<dim3(col_slabs, nsplit), THREADS1, 0, stream>>>(
        adj, fm, sacc, n_nodes);

    wmma_epilogue_kernel<<<1, 512, 0, stream>>>(sacc, A, B, out);
    (void)out_size; (void)ws_size; (void)n_in;
}